// myModel_35699768164470
// MI455X (gfx1250) — compile-verified
//
#include <hip/hip_runtime.h>

// ---------------------------------------------------------------------------
// CDNA5 (gfx1250) implementation of the VGG16+VQ forward pass.
// All large GEMMs (3x3 convs as 9 shifted GEMMs, whitening, 1x1 pre/post-VQ
// convs, sigma@sigma_inv_rac, FC head) use V_WMMA_F32_16X16X4_F32 so the
// math is fp32-exact w.r.t. the reference (only summation order differs).
// The conv kernel gives each wave a 64x16 output tile (4 accumulators) so a
// single B-fragment load feeds 4 WMMAs; all fragment loads are issued before
// the first WMMA of each step so they overlap the matrix pipe (staged waits).
// ---------------------------------------------------------------------------

typedef float v2f __attribute__((ext_vector_type(2)));
typedef float v8f __attribute__((ext_vector_type(8)));

#define WMMA_F32(a, b, c) __builtin_amdgcn_wmma_f32_16x16x4_f32( \
    false, (a), false, (b), (short)0, (c), false, false)

// ---------------------------------------------------------------------------
// 3x3 conv, pad=1, stride=1, NCHW fp32, computed as 9 shifted GEMMs:
//   out[:, n] += W[:, :, ky, kx] @ in_shifted(ky,kx)[:, n]
// GEMM view per tap: M=Cout, N=B*H*W, K=Cin.  Wave tile: 64(M) x 16(N).
// fp32 WMMA lane mapping (ISA 7.12.2):
//   A(m,k): lane = (m%16) + 16*(k>>1), elem = k&1
//   B(k,n): lane = (n%16) + 16*(k>>1), elem = k&1
//   D(m,n): lane = (n%16) + 16*h, vgpr v -> m = v + 8*h
// Out-of-image taps are zeroed by multiplying with vm in {0,1}; addresses are
// clamped in-bounds so the loads are always legal and finite (0*x exact).
// Requires Cout % 64 == 0 (holds for all 13 layers).
// ---------------------------------------------------------------------------
__global__ __launch_bounds__(32) void conv3x3_wmma_k(
    const float* __restrict__ in,   // [B][Cin][H][W]
    const float* __restrict__ w,    // [Cout][Cin][3][3]
    const float* __restrict__ bias, // [Cout]
    float* __restrict__ out,        // [B][Cout][H][W]
    int Cin, int H, int W, int Cout, int relu)
{
    const int lane = threadIdx.x;
    const int h = lane >> 4;          // lane half
    const int r = lane & 15;
    const int m0 = blockIdx.x * 64;   // Cout tile base (4 x 16 rows)
    const int n0 = blockIdx.y * 16;   // pixel tile base (n = b*H*W + y*W + x)
    const int HW = H * W;
    const int K9 = Cin * 9;
    const int ciL = 2 * h;            // lane's k-offset inside each 4-block
    const int CinMain = Cin & ~3;     // main-loop K extent (tail only for conv0)

    // this lane's output column (shared by B loads and D stores)
    const int n = n0 + r;
    const int b = n / HW;
    const int p = n % HW;
    const int y = p / W;
    const int x = p % W;
    const float* inb = in + (size_t)b * Cin * HW;
    const float* wr  = w + (size_t)(m0 + r) * K9;   // A row, m-tile 0
    const size_t mts = (size_t)16 * K9;             // m-tile stride in floats

    v8f acc0 = {0.f,0.f,0.f,0.f,0.f,0.f,0.f,0.f};
    v8f acc1 = acc0, acc2 = acc0, acc3 = acc0;

#pragma unroll
    for (int t = 0; t < 9; ++t) {
        const int dy = t / 3 - 1;     // compile-time after unroll
        const int dx = t % 3 - 1;
        const int yy = y + dy, xx = x + dx;
        const bool valid = ((unsigned)yy < (unsigned)H) && ((unsigned)xx < (unsigned)W);
        const float vm = valid ? 1.f : 0.f;       // hoisted per tap
        const int yyc = min(max(yy, 0), H - 1);   // clamped, always in-bounds
        const int xxc = min(max(xx, 0), W - 1);

        const float* ap0 = wr + ciL * 9 + t;      // A ptrs: +36 floats/iter
        const float* ap1 = ap0 + mts;
        const float* ap2 = ap1 + mts;
        const float* ap3 = ap2 + mts;
        const float* bp0 = inb + (size_t)yyc * W + xxc + (size_t)ciL * HW;
        const float* bp1 = bp0 + HW;              // B ptrs: +4*HW floats/iter

#pragma unroll 2
        for (int c0 = 0; c0 < CinMain; c0 += 4) {
            // ---- issue ALL loads first (distinct live values -> distinct
            //      VGPRs -> loads stay in flight across the WMMAs below) ----
            const float b0  = bp0[0], b1  = bp1[0];
            const float a00 = ap0[0], a01 = ap0[9];
            const float a10 = ap1[0], a11 = ap1[9];
            const float a20 = ap2[0], a21 = ap2[9];
            const float a30 = ap3[0], a31 = ap3[9];
            ap0 += 36; ap1 += 36; ap2 += 36; ap3 += 36;
            bp0 += (size_t)4 * HW;
            bp1 += (size_t)4 * HW;

            v2f bv;                               // shared by all 4 m-tiles
            bv[0] = b0 * vm;
            bv[1] = b1 * vm;
            v2f a;
            a[0] = a00; a[1] = a01; acc0 = WMMA_F32(a, bv, acc0);
            a[0] = a10; a[1] = a11; acc1 = WMMA_F32(a, bv, acc1);
            a[0] = a20; a[1] = a21; acc2 = WMMA_F32(a, bv, acc2);
            a[0] = a30; a[1] = a31; acc3 = WMMA_F32(a, bv, acc3);
        }

        if (CinMain < Cin) {          // uniform branch; only conv0 (Cin=3)
            const int ci0 = CinMain + ciL;
            const int ci1 = ci0 + 1;
            const size_t c0c = (size_t)min(ci0, Cin - 1);
            const size_t c1c = (size_t)min(ci1, Cin - 1);
            const float me0 = (ci0 < Cin) ? vm : 0.f;
            const float me1 = (ci1 < Cin) ? vm : 0.f;
            const float* bpt = inb + (size_t)yyc * W + xxc;
            const float* wt0 = wr + t;
            const float b0  = bpt[c0c * HW],           b1  = bpt[c1c * HW];
            const float a00 = wt0[c0c * 9],            a01 = wt0[c1c * 9];
            const float a10 = wt0[mts + c0c * 9],      a11 = wt0[mts + c1c * 9];
            const float a20 = wt0[2 * mts + c0c * 9],  a21 = wt0[2 * mts + c1c * 9];
            const float a30 = wt0[3 * mts + c0c * 9],  a31 = wt0[3 * mts + c1c * 9];
            v2f bv;
            bv[0] = b0 * me0;
            bv[1] = b1 * me1;
            v2f a;
            a[0] = a00; a[1] = a01; acc0 = WMMA_F32(a, bv, acc0);
            a[0] = a10; a[1] = a11; acc1 = WMMA_F32(a, bv, acc1);
            a[0] = a20; a[1] = a21; acc2 = WMMA_F32(a, bv, acc2);
            a[0] = a30; a[1] = a31; acc3 = WMMA_F32(a, bv, acc3);
        }
    }

    float* outb = out + (size_t)b * Cout * HW + p;
#pragma unroll
    for (int v = 0; v < 8; ++v) {
        const int mv = v + 8 * h;
        float v0 = acc0[v] + bias[m0 + mv];
        float v1 = acc1[v] + bias[m0 + 16 + mv];
        float v2 = acc2[v] + bias[m0 + 32 + mv];
        float v3 = acc3[v] + bias[m0 + 48 + mv];
        if (relu) {
            v0 = fmaxf(v0, 0.f); v1 = fmaxf(v1, 0.f);
            v2 = fmaxf(v2, 0.f); v3 = fmaxf(v3, 0.f);
        }
        outb[(size_t)(m0 + mv) * HW]      = v0;
        outb[(size_t)(m0 + 16 + mv) * HW] = v1;
        outb[(size_t)(m0 + 32 + mv) * HW] = v2;
        outb[(size_t)(m0 + 48 + mv) * HW] = v3;
    }
}

// ---------------------------------------------------------------------------
// Generic strided / batched-N fp32 WMMA GEMM (requires K % 4 == 0, ask even):
//   C(m,n) = sum_k A(m,k)*B(k,n) + bias_scale*bias[m]   (optional ReLU)
//   A(m,k) = A[m*ask + k]                      (k-contiguous -> b64 loads)
//   B(k,n) = Bm[(n/nper)*bbatch + k*bsk + (n%nper)*bsn]
//   C(m,n) = C[(n/nper)*cbatch + m*csm + (n%nper)*csn]
// The batched form expresses NCHW activations (nper = H*W) as well as flat
// row/column-major matrices (nper >= N, batch stride 0).
// ---------------------------------------------------------------------------
__global__ __launch_bounds__(32) void gemm_wmma_k(
    const float* __restrict__ A, int ask,
    const float* __restrict__ Bm, int nper, long long bbatch, int bsk, int bsn,
    const float* __restrict__ bias, float bscale,
    float* __restrict__ C, long long cbatch, int csm, int csn,
    int M, int N, int K, int relu)
{
    const int lane = threadIdx.x;
    const int h = lane >> 4;
    const int r = lane & 15;
    const int m0 = blockIdx.x * 16;
    const int n0 = blockIdx.y * 16;

    const int n = n0 + r;                     // N is always a multiple of 16
    const long long bB = (long long)(n / nper) * bbatch + (long long)(n % nper) * bsn;
    const long long cB = (long long)(n / nper) * cbatch + (long long)(n % nper) * csn;
    const int mA = min(m0 + r, M - 1);        // clamped: M-tail rows read row M-1,
                                              // their results are never stored
    const float* ap  = A + (size_t)mA * ask + 2 * h;   // 8B-aligned (ask even)
    const float* bp0 = Bm + bB + (long long)(2 * h) * bsk;
    const float* bp1 = bp0 + bsk;
    const long long bstep = 4LL * bsk;

    v8f acc = {0.f,0.f,0.f,0.f,0.f,0.f,0.f,0.f};

#pragma unroll 2
    for (int k0 = 0; k0 < K; k0 += 4) {
        const v2f a = *(const v2f*)ap;        // contiguous k-pair
        v2f bv;
        bv[0] = bp0[0];
        bv[1] = bp1[0];
        acc = WMMA_F32(a, bv, acc);
        ap  += 4;
        bp0 += bstep;
        bp1 += bstep;
    }

#pragma unroll
    for (int v = 0; v < 8; ++v) {
        const int m = m0 + v + 8 * h;
        if (m < M) {
            float val = acc[v];
            if (bias) val += bscale * bias[m];
            if (relu) val = fmaxf(val, 0.f);
            C[cB + (long long)m * csm] = val;
        }
    }
}

// ---------------------------------------------------------------------------
// 2x2 max-pool (stride 2) over [BC][H][W]; optional ReLU (ReLU commutes with
// max, so relu-then-pool == pool-then-relu).
// ---------------------------------------------------------------------------
__global__ void maxpool2_k(const float* __restrict__ in, float* __restrict__ out,
                           int BC, int H, int W, int relu)
{
    const int Ho = H >> 1, Wo = W >> 1;
    const int total = BC * Ho * Wo;
    int i = blockIdx.x * blockDim.x + threadIdx.x;
    if (i >= total) return;
    const int wo = i % Wo;
    int t = i / Wo;
    const int ho = t % Ho;
    const int bc = t / Ho;
    const float* p = in + ((size_t)bc * H + ho * 2) * W + wo * 2;
    float m = fmaxf(fmaxf(p[0], p[1]), fmaxf(p[W], p[W + 1]));
    if (relu) m = fmaxf(m, 0.f);
    out[i] = m;
}

// wv[d] = sum_c sir[d][c] * mu[c]   (folded into whitening bias with scale -1)
__global__ void matvec256_k(const float* __restrict__ Mt, const float* __restrict__ v,
                            float* __restrict__ out)
{
    const int d = threadIdx.x;   // 256 threads
    float s = 0.f;
    for (int c = 0; c < 256; ++c) s += Mt[d * 256 + c] * v[c];
    out[d] = s;
}

__global__ void zero_i32_k(int* p, int n)
{
    int i = blockIdx.x * blockDim.x + threadIdx.x;
    if (i < n) p[i] = 0;
}

__global__ void copy_f32_k(const float* __restrict__ a, float* __restrict__ b, int n)
{
    int i = blockIdx.x * blockDim.x + threadIdx.x;
    if (i < n) b[i] = a[i];
}

// ---------------------------------------------------------------------------
// VQ nearest-codebook assignment. 256 threads/block, one pixel vector (D=64)
// per thread. Codebook (512x64) is streamed through LDS in 64-row chunks.
// Also produces q[n][d]=E[idx][d], integer histogram counts, and per-block
// partial sums of ||q - z||^2 (fixed-order tree reduce -> deterministic).
// ---------------------------------------------------------------------------
#define VQ_CHUNK 64
__global__ __launch_bounds__(256) void vq_assign_k(
    const float* __restrict__ zf,   // [N][64]
    const float* __restrict__ E,    // [512][64]
    int* __restrict__ idx, float* __restrict__ q,
    int* __restrict__ counts, float* __restrict__ lpart)
{
    __shared__ float Es[VQ_CHUNK * 64];
    __shared__ float red[256];
    const int n = blockIdx.x * 256 + threadIdx.x;

    float zv[64];
#pragma unroll
    for (int d = 0; d < 64; ++d) zv[d] = zf[(size_t)n * 64 + d];

    float bestd = 3.4e38f;
    int best = 0;
    for (int c0 = 0; c0 < 512; c0 += VQ_CHUNK) {
        __syncthreads();
        for (int t = threadIdx.x; t < VQ_CHUNK * 64; t += 256)
            Es[t] = E[c0 * 64 + t];
        __syncthreads();
        for (int c = 0; c < VQ_CHUNK; ++c) {
            float s = 0.f;
#pragma unroll
            for (int d = 0; d < 64; ++d) {
                const float df = zv[d] - Es[c * 64 + d];
                s += df * df;
            }
            if (s < bestd) { bestd = s; best = c0 + c; }   // first-min (argmin)
        }
    }

    idx[n] = best;
    float lsum = 0.f;
#pragma unroll
    for (int d = 0; d < 64; ++d) {
        const float e = E[(size_t)best * 64 + d];
        q[(size_t)n * 64 + d] = e;
        const float df = e - zv[d];
        lsum += df * df;
    }
    atomicAdd(&counts[best], 1);    // integer: order-independent, deterministic

    red[threadIdx.x] = lsum;
    __syncthreads();
    for (int s = 128; s > 0; s >>= 1) {
        if ((int)threadIdx.x < s) red[threadIdx.x] += red[threadIdx.x + s];
        __syncthreads();
    }
    if (threadIdx.x == 0) lpart[blockIdx.x] = red[0];
}

__global__ void loss_reduce_k(const float* __restrict__ lpart, int n,
                              float* __restrict__ out, float scale)
{
    if (threadIdx.x == 0) {
        float s = 0.f;
        for (int i = 0; i < n; ++i) s += lpart[i];   // fixed order
        *out = s * scale;
    }
}

__global__ __launch_bounds__(512) void perplexity_k(const int* __restrict__ counts,
                                                    float* __restrict__ out, float invN)
{
    __shared__ float red[512];
    const int k = threadIdx.x;
    const float avg = (float)counts[k] * invN;
    red[k] = avg * logf(avg + 1e-10f);
    __syncthreads();
    for (int s = 256; s > 0; s >>= 1) {
        if (k < s) red[k] += red[k + s];
        __syncthreads();
    }
    if (k == 0) *out = expf(-red[0]);
}

// ---------------------------------------------------------------------------
// Host-side orchestration
// ---------------------------------------------------------------------------
extern "C" void kernel_launch(void* const* d_in, const int* in_sizes, int n_in,
                              void* d_out, int out_size, void* d_ws, size_t ws_size,
                              hipStream_t stream)
{
    (void)in_sizes; (void)n_in; (void)out_size; (void)ws_size;
    const int B = 256;

    const float* x     = (const float*)d_in[0];
    const float* mu    = (const float*)d_in[1];
    const float* sigma = (const float*)d_in[2];
    const float* sir   = (const float*)d_in[3];   // sigma_inv_rac
    auto convW = [&](int i) { return (const float*)d_in[4 + 2 * i]; };
    auto convB = [&](int i) { return (const float*)d_in[5 + 2 * i]; };
    const float* pre_w  = (const float*)d_in[30];
    const float* pre_b  = (const float*)d_in[31];
    const float* post_w = (const float*)d_in[32];
    const float* post_b = (const float*)d_in[33];
    const float* E      = (const float*)d_in[34];   // codebook [512][64]
    const float* fc1_w  = (const float*)d_in[35];
    const float* fc1_b  = (const float*)d_in[36];
    const float* fc2_w  = (const float*)d_in[37];
    const float* fc2_b  = (const float*)d_in[38];
    const float* fc3_w  = (const float*)d_in[39];
    const float* fc3_b  = (const float*)d_in[40];

    // ---- workspace layout (floats); requires ~145 MB of d_ws ----
    float* ws   = (float*)d_ws;
    float* actA = ws;                       // 16,777,216 f (67 MB)
    float* actB = actA + 16777216;          // 16,777,216 f
    float* S    = actB + 16777216;
    float* zf   = S;                        // 1,048,576 f  [16384][64]
    float* qb   = zf + 1048576;             // 1,048,576 f
    float* h1   = qb + 1048576;             // 131,072 f
    float* h2   = h1 + 131072;              // 131,072 f
    float* Mm   = h2 + 131072;              // 65,536 f (sigma @ sir)
    float* wv   = Mm + 65536;               // 256 f
    float* lpart= wv + 256;                 // 64 f
    int*   idxb = (int*)(lpart + 64);       // 16,384 i32
    int*   cnts = idxb + 16384;             // 512 i32

    // ---- output layout (floats, flat in reference return order) ----
    float* out        = (float*)d_out;
    float* out_loss   = out;                            // 1
    float* out_logits = out + 1;                        // 256*10
    float* out_perp   = out + 2561;                     // 1
    float* out_before = out + 2562;                     // [256][256][8][8]
    float* out_recon  = out_before + 4194304;           // [256][256][8][8]
    float* out_cb     = out_recon + 4194304;            // [512][64]

    auto conv = [&](const float* in, int i, int Cin, int H, int W, int Cout,
                    float* dst, int relu) {
        dim3 g(Cout / 64, (B * H * W) / 16);
        conv3x3_wmma_k<<<g, 32, 0, stream>>>(in, convW(i), convB(i), dst,
                                             Cin, H, W, Cout, relu);
    };
    auto pool = [&](const float* in, float* dst, int BC, int H, int W, int relu) {
        int total = BC * (H / 2) * (W / 2);
        maxpool2_k<<<(total + 255) / 256, 256, 0, stream>>>(in, dst, BC, H, W, relu);
    };
    auto gemm = [&](const float* A, int ask,
                    const float* Bm, int nper, long long bbatch, int bsk, int bsn,
                    const float* bias, float bscale,
                    float* C, long long cbatch, int csm, int csn,
                    int M, int N, int K, int relu) {
        dim3 g((M + 15) / 16, (N + 15) / 16);
        gemm_wmma_k<<<g, 32, 0, stream>>>(A, ask, Bm, nper, bbatch, bsk, bsn,
                                          bias, bscale, C, cbatch, csm, csn,
                                          M, N, K, relu);
    };

    // ---- VGG trunk up to the VQ insertion point ----
    conv(x,    0,   3, 32, 32,  64, actA, 1);
    conv(actA, 1,  64, 32, 32,  64, actB, 1);
    pool(actB, actA, B * 64, 32, 32, 0);
    conv(actA, 2,  64, 16, 16, 128, actB, 1);
    conv(actB, 3, 128, 16, 16, 128, actA, 1);
    pool(actA, actB, B * 128, 16, 16, 0);
    conv(actB, 4, 128,  8,  8, 256, actA, 1);
    conv(actA, 5, 256,  8,  8, 256, actB, 1);
    conv(actB, 6, 256,  8,  8, 256, out_before, 0);   // data_before (pre-ReLU)

    // ---- VQ compression block (N = 256*8*8 = 16384 pixel vectors) ----
    // whitening: xc = sir @ (x - mu) = sir @ x - (sir @ mu)
    matvec256_k<<<1, 256, 0, stream>>>(sir, mu, wv);
    gemm(sir, 256, out_before, 64, 16384, 64, 1, wv, -1.f,
         actA, 16384, 64, 1, 256, 16384, 256, 0);
    // z = pre_w @ xc + pre_b   -> zf [N][64]
    gemm(pre_w, 256, actA, 64, 16384, 64, 1, pre_b, 1.f,
         zf, 4096, 1, 64, 64, 16384, 256, 0);
    // nearest-code assignment + q + counts + loss partials
    zero_i32_k<<<2, 256, 0, stream>>>(cnts, 512);
    vq_assign_k<<<64, 256, 0, stream>>>(zf, E, idxb, qb, cnts, lpart);
    loss_reduce_k<<<1, 32, 0, stream>>>(lpart, 64, out_loss,
                                        0.25f / (16384.f * 64.f));
    perplexity_k<<<1, 512, 0, stream>>>(cnts, out_perp, 1.f / 16384.f);
    // M = sigma @ sigma_inv_rac
    gemm(sigma, 256, sir, 256, 0, 256, 1, nullptr, 0.f,
         Mm, 0, 256, 1, 256, 256, 256, 0);
    // xr = post_w @ q + post_b   -> actB [B][256][8][8]
    gemm(post_w, 64, qb, 64, 4096, 1, 64, post_b, 1.f,
         actB, 16384, 64, 1, 256, 16384, 64, 0);
    // data_recon = M @ xr + mu
    gemm(Mm, 256, actB, 64, 16384, 64, 1, mu, 1.f,
         out_recon, 16384, 64, 1, 256, 16384, 256, 0);
    // relu + pool (ReLU fused into pool since both are monotone maxes)
    pool(out_recon, actA, B * 256, 8, 8, 1);

    // ---- remainder of the trunk ----
    conv(actA,  7, 256, 4, 4, 512, actB, 1);
    conv(actB,  8, 512, 4, 4, 512, actA, 1);
    conv(actA,  9, 512, 4, 4, 512, actB, 1);
    pool(actB, actA, B * 512, 4, 4, 0);
    conv(actA, 10, 512, 2, 2, 512, actB, 1);
    conv(actB, 11, 512, 2, 2, 512, actA, 1);
    conv(actA, 12, 512, 2, 2, 512, actB, 1);
    pool(actB, actA, B * 512, 2, 2, 0);   // -> actA = [B][512] (flatten)

    // ---- FC head: h = relu(x @ W^T + b) computed as W @ x^T ----
    gemm(fc1_w, 512, actA, 256, 0, 1, 512, fc1_b, 1.f,
         h1, 0, 1, 512, 512, 256, 512, 1);
    gemm(fc2_w, 512, h1, 256, 0, 1, 512, fc2_b, 1.f,
         h2, 0, 1, 512, 512, 256, 512, 1);
    gemm(fc3_w, 512, h2, 256, 0, 1, 512, fc3_b, 1.f,
         out_logits, 0, 1, 10, 10, 256, 512, 0);

    // ---- codebook passthrough output ----
    copy_f32_k<<<(32768 + 255) / 256, 256, 0, stream>>>(E, out_cb, 32768);
}